// FourierInterpolator_83296595739091
// MI455X (gfx1250) — compile-verified
//
#include <hip/hip_runtime.h>

typedef __attribute__((ext_vector_type(2))) float v2f;
typedef __attribute__((ext_vector_type(8))) float v8f;

#define N_GRID 1024
#define P_TOTAL 4096
#define NBATCH 8
static constexpr float TWO_PI = 6.28318530717958647692f;

// ---------------------------------------------------------------------------
// Kernel 0: exact 2D DFT of y (8 batches of 32x32) via two 1D passes in LDS.
// Twiddles come from a 32-entry table (phase is integer mod 32), so the inner
// loops are pure FMA. Output: ReF/ImF row-major [kx][ky] into workspace.
// ---------------------------------------------------------------------------
__global__ __launch_bounds__(256) void dft2d_kernel(const float* __restrict__ y,
                                                    float* __restrict__ wsRe,
                                                    float* __restrict__ wsIm) {
  __shared__ float ly[1024];
  __shared__ float tRe[1024];
  __shared__ float tIm[1024];
  __shared__ float tabC[32];
  __shared__ float tabS[32];
  const int b = blockIdx.x;
  const int tid = threadIdx.x;

  for (int i = tid; i < 1024; i += 256) ly[i] = y[b * 1024 + i];
  if (tid < 32) {
    float s, c;
    __sincosf((TWO_PI / 32.0f) * (float)tid, &s, &c);
    tabC[tid] = c;
    tabS[tid] = s;
  }
  __syncthreads();

  // Pass 1: T[kx][ny] = sum_nx y[nx][ny] * exp(-i*2pi*kx*nx/32)
  for (int o = tid; o < 1024; o += 256) {
    const int kx = o >> 5, ny = o & 31;
    float re = 0.f, im = 0.f;
#pragma unroll 4
    for (int nx = 0; nx < 32; ++nx) {
      const int idx = (kx * nx) & 31;
      const float v = ly[nx * 32 + ny];
      re = __builtin_fmaf(v, tabC[idx], re);
      im = __builtin_fmaf(-v, tabS[idx], im);
    }
    tRe[o] = re;  // o = kx*32 + ny
    tIm[o] = im;
  }
  __syncthreads();

  // Pass 2: F[kx][ky] = sum_ny T[kx][ny] * exp(-i*2pi*ky*ny/32)
  for (int o = tid; o < 1024; o += 256) {
    const int kx = o >> 5, ky = o & 31;
    float re = 0.f, im = 0.f;
#pragma unroll 4
    for (int ny = 0; ny < 32; ++ny) {
      const int idx = (ky * ny) & 31;
      const float tr = tRe[kx * 32 + ny];
      const float ti = tIm[kx * 32 + ny];
      const float c = tabC[idx], s = tabS[idx];
      re += tr * c + ti * s;
      im += ti * c - tr * s;
    }
    wsRe[b * 1024 + o] = re;
    wsIm[b * 1024 + o] = im;
  }
}

// ---------------------------------------------------------------------------
// Kernel 1: factorized synthesis.
//   A [32xP] = ReF@Cy + ImF@(-Sy),  Bm[32xP] = ReF@Sy + ImF@Cy   (WMMA f32)
//   out[p]   = (1/1024) * sum_kx ( cx[kx,p]*A[kx,p] - sx[kx,p]*Bm[kx,p] )
// One wave per 16-point tile; block = 8 waves (128 points) for one batch.
// ---------------------------------------------------------------------------
__global__ __launch_bounds__(256) void synth_kernel(const float* __restrict__ xnew,
                                                    const float* __restrict__ wsRe,
                                                    const float* __restrict__ wsIm,
                                                    float* __restrict__ out) {
  __shared__ float lRe[1024];
  __shared__ float lIm[1024];
  const int b = blockIdx.y;
  const int tid = threadIdx.x;

  for (int i = tid; i < 1024; i += 256) {
    lRe[i] = wsRe[b * 1024 + i];
    lIm[i] = wsIm[b * 1024 + i];
  }
  __syncthreads();

  const int wave = tid >> 5;   // 0..7 tiles per block
  const int lane = tid & 31;
  const int g = lane >> 4;     // half-wave group (K/M split per WMMA layout)
  const int n = lane & 15;     // column (point) index within the 16-point tile
  const int p = blockIdx.x * 128 + wave * 16 + n;  // global point index

  // Per-lane base angles for this lane's point (lanes 16..31 mirror 0..15).
  const size_t xbase = ((size_t)b * P_TOTAL + (size_t)p) * 2;
  const float ax = xnew[xbase + 0] * 32.0f * (TWO_PI / 32.0f);
  const float ay = xnew[xbase + 1] * 32.0f * (TWO_PI / 32.0f);

  // y-dim harmonics, packed for the 4x16 f32 B layout:
  //   B fragment for kstep ks = { B[4ks+2g][n], B[4ks+2g+1][n] }
  float cy[16], sy[16];
#pragma unroll
  for (int ks = 0; ks < 8; ++ks) {
#pragma unroll
    for (int j = 0; j < 2; ++j) {
      const float kf = (float)(4 * ks + 2 * g + j);
      float s, c;
      __sincosf(kf * ay, &s, &c);
      cy[2 * ks + j] = c;
      sy[2 * ks + j] = s;
    }
  }
  // x-dim harmonics mapped to the C/D accumulator layout:
  //   VGPR v of tile mt holds kx = 16*mt + 8*g + v for this lane.
  float cx[16], sx[16];
#pragma unroll
  for (int i = 0; i < 16; ++i) {
    const int mt = i >> 3, v = i & 7;
    const float kf = (float)(16 * mt + 8 * g + v);
    float s, c;
    __sincosf(kf * ax, &s, &c);
    cx[i] = c;
    sx[i] = s;
  }

  v8f accA0 = {}, accA1 = {}, accB0 = {}, accB1 = {};
  const int m = n;  // A-matrix row held by this lane (same for both halves)

#pragma unroll
  for (int ks = 0; ks < 8; ++ks) {
    const int c0 = 4 * ks + 2 * g;  // even -> 8B-aligned LDS loads
    const v2f bcy = {cy[2 * ks], cy[2 * ks + 1]};
    const v2f bsy = {sy[2 * ks], sy[2 * ks + 1]};
    const v2f bnsy = {-sy[2 * ks], -sy[2 * ks + 1]};

    // M-tile 0: kx rows 0..15
    const v2f re0 = *(const v2f*)&lRe[m * 32 + c0];
    const v2f im0 = *(const v2f*)&lIm[m * 32 + c0];
    accA0 = __builtin_amdgcn_wmma_f32_16x16x4_f32(false, re0, false, bcy,  (short)0, accA0, false, false);
    accA0 = __builtin_amdgcn_wmma_f32_16x16x4_f32(false, im0, false, bnsy, (short)0, accA0, false, false);
    accB0 = __builtin_amdgcn_wmma_f32_16x16x4_f32(false, re0, false, bsy,  (short)0, accB0, false, false);
    accB0 = __builtin_amdgcn_wmma_f32_16x16x4_f32(false, im0, false, bcy,  (short)0, accB0, false, false);

    // M-tile 1: kx rows 16..31
    const v2f re1 = *(const v2f*)&lRe[(m + 16) * 32 + c0];
    const v2f im1 = *(const v2f*)&lIm[(m + 16) * 32 + c0];
    accA1 = __builtin_amdgcn_wmma_f32_16x16x4_f32(false, re1, false, bcy,  (short)0, accA1, false, false);
    accA1 = __builtin_amdgcn_wmma_f32_16x16x4_f32(false, im1, false, bnsy, (short)0, accA1, false, false);
    accB1 = __builtin_amdgcn_wmma_f32_16x16x4_f32(false, re1, false, bsy,  (short)0, accB1, false, false);
    accB1 = __builtin_amdgcn_wmma_f32_16x16x4_f32(false, im1, false, bcy,  (short)0, accB1, false, false);
  }

  // Epilogue: contract over kx directly on the accumulator layout.
  float partial = 0.f;
#pragma unroll
  for (int v = 0; v < 8; ++v) {
    partial += cx[v] * accA0[v] - sx[v] * accB0[v];
    partial += cx[8 + v] * accA1[v] - sx[8 + v] * accB1[v];
  }
  // Lane L and L+16 hold the same point p with complementary kx sets.
  partial += __shfl_xor(partial, 16, 32);

  if (lane < 16) {
    out[(size_t)b * P_TOTAL + (size_t)p] = partial * (1.0f / 1024.0f);
  }
}

// ---------------------------------------------------------------------------
// Launcher
// ---------------------------------------------------------------------------
extern "C" void kernel_launch(void* const* d_in, const int* in_sizes, int n_in,
                              void* d_out, int out_size, void* d_ws, size_t ws_size,
                              hipStream_t stream) {
  const float* y = (const float*)d_in[0];     // (8, 32, 32) f32
  const float* xnew = (const float*)d_in[1];  // (8, 64, 64, 2) f32
  float* out = (float*)d_out;                 // (8, 64, 64) f32

  float* wsRe = (float*)d_ws;                 // 8*1024 floats
  float* wsIm = wsRe + (size_t)NBATCH * N_GRID;

  dft2d_kernel<<<NBATCH, 256, 0, stream>>>(y, wsRe, wsIm);
  synth_kernel<<<dim3(P_TOTAL / 128, NBATCH), 256, 0, stream>>>(xnew, wsRe, wsIm, out);
}